// PairwiseDiffLoss_39513699123824
// MI455X (gfx1250) — compile-verified
//
#include <hip/hip_runtime.h>
#include <hip/hip_bf16.h>

// CDNA5 / gfx1250, wave32.
typedef __attribute__((ext_vector_type(2))) float v2f;
typedef __attribute__((ext_vector_type(8))) float v8f;

#define N_ITEMS 1024
#define TPB     256
#define PER_T   (N_ITEMS / TPB)   // 4 elements per thread

// One workgroup (256 threads = 8 wave32) per batch row.
// Computes total_b (pre-normalized by Z) and count_b, stores to tc[2b], tc[2b+1].
__global__ void __launch_bounds__(TPB)
pdl_batch_kernel(const float* __restrict__ scores,
                 const int*   __restrict__ rel,
                 float*       __restrict__ tc) {
    __shared__ float red[TPB];
    __shared__ float acc[10];   // acc[0..4] = S[g] (exp sums), acc[5..9] = h[g] (counts)

    const int b = blockIdx.x;
    const int t = threadIdx.x;
    const float* srow = scores + (size_t)b * N_ITEMS;
    const int*   rrow = rel    + (size_t)b * N_ITEMS;

    // ---- load 4 strided elements (coalesced), local max ----
    float s[PER_T];
    int   r[PER_T];
#pragma unroll
    for (int k = 0; k < PER_T; ++k) {
        s[k] = srow[t + k * TPB];
        r[k] = rrow[t + k * TPB];
    }
    float m = fmaxf(fmaxf(s[0], s[1]), fmaxf(s[2], s[3]));

    // ---- block max reduction (numerically stable softmax) ----
    red[t] = m;
    __syncthreads();
#pragma unroll
    for (int off = TPB / 2; off > 0; off >>= 1) {
        if (t < off) red[t] = fmaxf(red[t], red[t + off]);
        __syncthreads();
    }
    const float rowmax = red[0];

    if (t < 10) acc[t] = 0.0f;
    __syncthreads();

    // ---- per-grade exp sums + histogram via LDS atomics (ds_add_f32) ----
#pragma unroll
    for (int k = 0; k < PER_T; ++k) {
        float e = __expf(s[k] - rowmax);
        atomicAdd(&acc[r[k]],     e);
        atomicAdd(&acc[5 + r[k]], 1.0f);
    }
    __syncthreads();

    // ---- 5x5 grade interaction as two f32 outer products on the WMMA unit ----
    // D_S = S * h^T, D_h = h * h^T  via V_WMMA_F32_16X16X4_F32 (K=0 column/row only).
    // Wave-uniform path: every wave executes with full EXEC (WMMA requirement).
    const int lane = t & 31;
    const float Sv = (lane < 5) ? acc[lane]     : 0.0f;
    const float hv = (lane < 5) ? acc[5 + lane] : 0.0f;

    // A 16x4 f32: VGPR0 = K=0 (lanes 0-15) / K=2 (lanes 16-31); VGPR1 = K=1/K=3.
    v2f aS; aS.x = Sv; aS.y = 0.0f;     // only column K=0 nonzero; lanes>=5 already 0
    v2f aH; aH.x = hv; aH.y = 0.0f;
    // B 4x16 f32: row K=0 lives in lanes 0-15 of the first VGPR; rest zeroed.
    v2f bH; bH.x = (lane < 16) ? hv : 0.0f; bH.y = 0.0f;
    v8f c0 = {};

    v8f dS = __builtin_amdgcn_wmma_f32_16x16x4_f32(
        false, aS, false, bH, (short)0, c0, false, false);
    v8f dH = __builtin_amdgcn_wmma_f32_16x16x4_f32(
        false, aH, false, bH, (short)0, c0, false, false);

    // C/D layout: lanes 0-15 -> n=lane, m=v ; lanes 16-31 -> n=lane-16, m=v+8.
    // total = sum_{m>n} (-D_S[m][n]) + sum_{m<n} (+D_S[m][n]) ; count = sum_{m>n} D_h[m][n]
    const int n  = lane & 15;
    const int mb = (lane >> 4) * 8;
    float pt = 0.0f, pc = 0.0f;
#pragma unroll
    for (int v = 0; v < 8; ++v) {
        const int mm = mb + v;
        const float ds = dS[v];
        pt += (mm > n) ? -ds : ((mm < n) ? ds : 0.0f);
        pc += (mm > n) ? dH[v] : 0.0f;
    }
    // wave32 butterfly reduction
#pragma unroll
    for (int off = 16; off > 0; off >>= 1) {
        pt += __shfl_xor(pt, off, 32);
        pc += __shfl_xor(pc, off, 32);
    }

    if (t == 0) {
        const float Z = acc[0] + acc[1] + acc[2] + acc[3] + acc[4]; // softmax denom
        tc[2 * b]     = pt / Z;   // total_b (softmax-normalized)
        tc[2 * b + 1] = pc;       // count_b
    }
}

// Final reduction over B batches: out = sum(total) / sum(count)
__global__ void __launch_bounds__(128)
pdl_final_kernel(const float* __restrict__ tc, float* __restrict__ out, int B) {
    __shared__ float st[128], sc[128];
    const int t = threadIdx.x;
    float a = 0.0f, c = 0.0f;
    for (int i = t; i < B; i += 128) {
        a += tc[2 * i];
        c += tc[2 * i + 1];
    }
    st[t] = a; sc[t] = c;
    __syncthreads();
#pragma unroll
    for (int off = 64; off > 0; off >>= 1) {
        if (t < off) { st[t] += st[t + off]; sc[t] += sc[t + off]; }
        __syncthreads();
    }
    if (t == 0) out[0] = st[0] / sc[0];
}

extern "C" void kernel_launch(void* const* d_in, const int* in_sizes, int n_in,
                              void* d_out, int out_size, void* d_ws, size_t ws_size,
                              hipStream_t stream) {
    const float* scores = (const float*)d_in[0];
    const int*   rel    = (const int*)d_in[1];
    float*       out    = (float*)d_out;
    float*       tc     = (float*)d_ws;   // 2*B floats of scratch

    const int B = in_sizes[0] / N_ITEMS;

    pdl_batch_kernel<<<B, TPB, 0, stream>>>(scores, rel, tc);
    pdl_final_kernel<<<1, 128, 0, stream>>>(tc, out, B);
}